// Net_12781822673245
// MI455X (gfx1250) — compile-verified
//
#include <hip/hip_runtime.h>
#include <math.h>

#define N_NODES   100000
#define N_EDGES   3200000
#define N_FEAT    512
#define HIDDEN    16
#define N_CLASSES 40
#define NPAD      48            // 3 WMMA n-tiles of 16
#define ROW_TILES (N_NODES / 16)  // 6250 exactly

typedef __attribute__((ext_vector_type(2))) float v2f;
typedef __attribute__((ext_vector_type(8))) float v8f;

// ---------------- degree / normalization ----------------

__global__ void k_init_deg(float* __restrict__ deg) {
    int n = blockIdx.x * blockDim.x + threadIdx.x;
    if (n < N_NODES) deg[n] = 1.0f;               // self-loop weight 1
}

__global__ void k_accum_deg(const int* __restrict__ ei,
                            const float* __restrict__ w,
                            float* __restrict__ deg) {
    int e = blockIdx.x * blockDim.x + threadIdx.x;
    if (e < N_EDGES) atomicAdd(&deg[ei[N_EDGES + e]], w[e]);   // dst row
}

__global__ void k_dinv(const float* __restrict__ deg, float* __restrict__ dinv) {
    int n = blockIdx.x * blockDim.x + threadIdx.x;
    if (n < N_NODES) dinv[n] = rsqrtf(deg[n]);
}

// ---------------- layer 1 GEMM: XW = X @ W1  (100000x512 @ 512x16) ----------------

__global__ __launch_bounds__(256) void k_gemm_xw1(const float* __restrict__ X,
                                                  const float* __restrict__ W1,
                                                  float* __restrict__ XW) {
    __shared__ float lw[N_FEAT * HIDDEN];   // 32 KB of LDS: full W1
    {
        const float4* s4 = reinterpret_cast<const float4*>(W1);
        float4* d4 = reinterpret_cast<float4*>(lw);
        for (int i = threadIdx.x; i < (N_FEAT * HIDDEN) / 4; i += blockDim.x) d4[i] = s4[i];
    }
    __syncthreads();

    const int wave = threadIdx.x >> 5;
    const int lane = threadIdx.x & 31;
    const int tile = blockIdx.x * 8 + wave;      // one 16-row tile per wave
    if (tile >= ROW_TILES) return;               // wave-uniform: EXEC stays all-ones

    const int col   = lane & 15;                 // N index / M index depending on role
    const int khalf = (lane >> 4) * 2;           // lanes 0-15: K+0/1, lanes 16-31: K+2/3
    const float* xrow = X + (long long)(tile * 16 + col) * N_FEAT;

    v8f acc = {};
    for (int k = 0; k < N_FEAT; k += 4) {
        // A fragment 16x4 f32: lane holds (row=col, K=k+khalf, k+khalf+1) -> contiguous b64 load
        v2f a = *reinterpret_cast<const v2f*>(xrow + k + khalf);
        // B fragment 4x16 f32: mirrored layout, rows from LDS
        v2f b;
        b.x = lw[(k + khalf) * HIDDEN + col];
        b.y = lw[(k + khalf + 1) * HIDDEN + col];
        acc = __builtin_amdgcn_wmma_f32_16x16x4_f32(false, a, false, b,
                                                    (short)0, acc, false, false);
    }

    // D layout: VGPR r -> (M=r, N=lane) for lanes 0-15 ; (M=r+8, N=lane-16) for 16-31
    const int mbase = tile * 16 + ((lane >> 4) ? 8 : 0);
#pragma unroll
    for (int r = 0; r < 8; ++r)
        XW[(long long)(mbase + r) * HIDDEN + col] = acc[r];
}

// ---------------- layer 1 aggregation ----------------

__global__ void k_selfloop16(const float* __restrict__ xw,
                             const float* __restrict__ dinv,
                             float* __restrict__ h1) {
    int i = blockIdx.x * blockDim.x + threadIdx.x;   // N_NODES*16 = 1.6M
    if (i < N_NODES * HIDDEN) {
        float di = dinv[i >> 4];
        h1[i] = xw[i] * (di * di);                   // xw / deg
    }
}

__global__ void k_edge16(const int* __restrict__ ei,
                         const float* __restrict__ w,
                         const float* __restrict__ dinv,
                         const float* __restrict__ xw,
                         float* __restrict__ h1) {
    unsigned t = blockIdx.x * blockDim.x + threadIdx.x;    // E*16 = 51.2M
    if (t >= (unsigned)N_EDGES * HIDDEN) return;
    int e = (int)(t >> 4), c = (int)(t & 15);
    int s = ei[e], d = ei[N_EDGES + e];
    float norm = dinv[s] * w[e] * dinv[d];
    atomicAdd(&h1[(long long)d * HIDDEN + c], xw[(long long)s * HIDDEN + c] * norm);
}

__global__ void k_relu_bias(const float* __restrict__ b1, float* __restrict__ h1) {
    int i = blockIdx.x * blockDim.x + threadIdx.x;
    if (i < N_NODES * HIDDEN) h1[i] = fmaxf(h1[i] + b1[i & 15], 0.0f);
}

// ---------------- layer 2 GEMM: HW = H @ W2  (100000x16 @ 16x40, N padded to 48) ----------------

__global__ __launch_bounds__(256) void k_gemm_hw2(const float* __restrict__ H,
                                                  const float* __restrict__ W2,
                                                  float* __restrict__ HW) {
    __shared__ float lw[HIDDEN * NPAD];              // 16x48 zero-padded W2
    for (int i = threadIdx.x; i < HIDDEN * NPAD; i += blockDim.x) {
        int k = i / NPAD, n = i % NPAD;
        lw[i] = (n < N_CLASSES) ? W2[k * N_CLASSES + n] : 0.0f;
    }
    __syncthreads();

    const int wave = threadIdx.x >> 5;
    const int lane = threadIdx.x & 31;
    const int tile = blockIdx.x * 8 + wave;
    if (tile >= ROW_TILES) return;

    const int col   = lane & 15;
    const int khalf = (lane >> 4) * 2;
    const float* hrow = H + (long long)(tile * 16 + col) * HIDDEN;

    v8f c0 = {}, c1 = {}, c2 = {};
#pragma unroll
    for (int k = 0; k < HIDDEN; k += 4) {
        v2f a = *reinterpret_cast<const v2f*>(hrow + k + khalf);  // reused for 3 n-tiles
        int kr = k + khalf;
        v2f b0, b1v, b2v;
        b0.x  = lw[kr * NPAD + col];       b0.y  = lw[(kr + 1) * NPAD + col];
        b1v.x = lw[kr * NPAD + 16 + col];  b1v.y = lw[(kr + 1) * NPAD + 16 + col];
        b2v.x = lw[kr * NPAD + 32 + col];  b2v.y = lw[(kr + 1) * NPAD + 32 + col];
        c0 = __builtin_amdgcn_wmma_f32_16x16x4_f32(false, a, false, b0,  (short)0, c0, false, false);
        c1 = __builtin_amdgcn_wmma_f32_16x16x4_f32(false, a, false, b1v, (short)0, c1, false, false);
        c2 = __builtin_amdgcn_wmma_f32_16x16x4_f32(false, a, false, b2v, (short)0, c2, false, false);
    }

    const int mbase = tile * 16 + ((lane >> 4) ? 8 : 0);
#pragma unroll
    for (int r = 0; r < 8; ++r) {
        long long base = (long long)(mbase + r) * N_CLASSES;
        HW[base + col]      = c0[r];
        HW[base + 16 + col] = c1[r];
        if (col < 8) HW[base + 32 + col] = c2[r];    // drop padded cols 40..47
    }
}

// ---------------- layer 2 aggregation + log_softmax ----------------

__global__ void k_selfloop40(const float* __restrict__ hw,
                             const float* __restrict__ dinv,
                             float* __restrict__ out) {
    unsigned i = blockIdx.x * blockDim.x + threadIdx.x;   // N*40 = 4M
    if (i < (unsigned)N_NODES * N_CLASSES) {
        float di = dinv[i / N_CLASSES];
        out[i] = hw[i] * (di * di);
    }
}

__global__ void k_edge40(const int* __restrict__ ei,
                         const float* __restrict__ w,
                         const float* __restrict__ dinv,
                         const float* __restrict__ hw,
                         float* __restrict__ out) {
    unsigned t = blockIdx.x * blockDim.x + threadIdx.x;   // E*40 = 128M
    if (t >= (unsigned)N_EDGES * N_CLASSES) return;
    int e = (int)(t / N_CLASSES), c = (int)(t % N_CLASSES);
    int s = ei[e], d = ei[N_EDGES + e];
    float norm = dinv[s] * w[e] * dinv[d];
    atomicAdd(&out[(long long)d * N_CLASSES + c], hw[(long long)s * N_CLASSES + c] * norm);
}

__global__ void k_logsoftmax(const float* __restrict__ b2, float* __restrict__ out) {
    int n = blockIdx.x * blockDim.x + threadIdx.x;
    if (n >= N_NODES) return;
    float* row = out + (long long)n * N_CLASSES;
    float z[N_CLASSES];
    float m = -INFINITY;
#pragma unroll
    for (int c = 0; c < N_CLASSES; ++c) { z[c] = row[c] + b2[c]; m = fmaxf(m, z[c]); }
    float s = 0.0f;
#pragma unroll
    for (int c = 0; c < N_CLASSES; ++c) s += __expf(z[c] - m);
    float lse = m + __logf(s);
#pragma unroll
    for (int c = 0; c < N_CLASSES; ++c) row[c] = z[c] - lse;
}

// ---------------- launcher ----------------

extern "C" void kernel_launch(void* const* d_in, const int* in_sizes, int n_in,
                              void* d_out, int out_size, void* d_ws, size_t ws_size,
                              hipStream_t stream) {
    const float* x  = (const float*)d_in[0];
    const int*   ei = (const int*)  d_in[1];
    const float* w  = (const float*)d_in[2];
    const float* W1 = (const float*)d_in[3];
    const float* b1 = (const float*)d_in[4];
    const float* W2 = (const float*)d_in[5];
    const float* b2 = (const float*)d_in[6];
    float* out = (float*)d_out;

    float* ws   = (float*)d_ws;
    float* deg  = ws;                        // 100000
    float* dinv = deg + N_NODES;             // 100000
    float* xw   = dinv + N_NODES;            // 1.6M
    float* h1   = xw + (size_t)N_NODES * HIDDEN;   // 1.6M
    float* hw2  = h1 + (size_t)N_NODES * HIDDEN;   // 4.0M  (~29.6 MB total)

    const int B = 256;
    // normalization
    k_init_deg <<<(N_NODES + B - 1) / B, B, 0, stream>>>(deg);
    k_accum_deg<<<(N_EDGES + B - 1) / B, B, 0, stream>>>(ei, w, deg);
    k_dinv     <<<(N_NODES + B - 1) / B, B, 0, stream>>>(deg, dinv);
    // layer 1
    k_gemm_xw1 <<<(ROW_TILES + 7) / 8, B, 0, stream>>>(x, W1, xw);
    k_selfloop16<<<(N_NODES * HIDDEN + B - 1) / B, B, 0, stream>>>(xw, dinv, h1);
    k_edge16   <<<((unsigned)N_EDGES * HIDDEN + B - 1) / B, B, 0, stream>>>(ei, w, dinv, xw, h1);
    k_relu_bias<<<(N_NODES * HIDDEN + B - 1) / B, B, 0, stream>>>(b1, h1);
    // layer 2
    k_gemm_hw2 <<<(ROW_TILES + 7) / 8, B, 0, stream>>>(h1, W2, hw2);
    k_selfloop40<<<((unsigned)N_NODES * N_CLASSES + B - 1) / B, B, 0, stream>>>(hw2, dinv, out);
    k_edge40   <<<((unsigned)N_EDGES * N_CLASSES + B - 1) / B, B, 0, stream>>>(ei, w, dinv, hw2, out);
    k_logsoftmax<<<(N_NODES + B - 1) / B, B, 0, stream>>>(b2, out);
}